// PointNet_89541478187052
// MI455X (gfx1250) — compile-verified
//
#include <hip/hip_runtime.h>

typedef __attribute__((ext_vector_type(2))) float v2f;
typedef __attribute__((ext_vector_type(8))) float v8f;

#define WAVES 8
#define BLOCK 256

__device__ __forceinline__ v8f wmma4(v2f a, v2f b, v8f c) {
  // D = A(16x4) * B(4x16) + C(16x16), fp32 WMMA
  return __builtin_amdgcn_wmma_f32_16x16x4_f32(false, a, false, b, (short)0, c,
                                               false, false);
}

// ---------------------------------------------------------------------------
// Pass A: per-edge x = ef @ W1 + b1 (WMMA), accumulate per-channel sum/sumsq
// for BatchNorm statistics. Nothing else stored (x is recomputed in pass C).
// ---------------------------------------------------------------------------
template <int KP, int KS, bool SECOND>
__global__ __launch_bounds__(BLOCK) void edge_pass_a(
    const float* __restrict__ pos, const int* __restrict__ esrc,
    const int* __restrict__ edst, const float* __restrict__ hprev,
    const float* __restrict__ W1, const float* __restrict__ b1,
    float* __restrict__ gSum, float* __restrict__ gSq, int E) {
  __shared__ float w1t[64][KS];          // transposed weights, zero K-pad
  __shared__ float ef[WAVES][16][KS];    // per-wave 16-edge feature tile
  __shared__ float b1s[64];
  __shared__ float sSum[64], sSq[64];

  const int tid = threadIdx.x;
  const int w = tid >> 5;
  const int lane = tid & 31;
  const int m = lane & 15;
  const int hi = lane >> 4;
  const int koff = hi * 2;
  const int KIN = SECOND ? 67 : 6;

  for (int idx = tid; idx < 64 * KS; idx += BLOCK) {
    int n = idx / KS, k = idx - n * KS;
    w1t[n][k] = (k < KIN) ? W1[k * 64 + n] : 0.0f;
  }
  if (tid < 64) {
    b1s[tid] = b1[tid];
    sSum[tid] = 0.0f;
    sSq[tid] = 0.0f;
  }

  const int base = (blockIdx.x * WAVES + w) * 16;
  {
    int e = base + m;
    if (e >= E) e = E - 1;
    int s = esrc[e];
    if (SECOND) {  // cols 0..63 = h_prev[src]  (L2-resident gather)
      const float4* hr = (const float4*)(hprev + (size_t)s * 64);
      float4* d4 = (float4*)&ef[w][m][hi * 32];
#pragma unroll
      for (int i = 0; i < 8; ++i) d4[i] = hr[hi * 8 + i];
    }
    if (lane < 16) {
      int d = edst[e];
      float sx = pos[3 * s], sy = pos[3 * s + 1], sz = pos[3 * s + 2];
      float dx = pos[3 * d], dy = pos[3 * d + 1], dz = pos[3 * d + 2];
      if (SECOND) {
        ef[w][m][64] = sx - dx; ef[w][m][65] = sy - dy;
        ef[w][m][66] = sz - dz; ef[w][m][67] = 0.0f;
      } else {
        ef[w][m][0] = sx; ef[w][m][1] = sy; ef[w][m][2] = sz;
        ef[w][m][3] = sx - dx; ef[w][m][4] = sy - dy; ef[w][m][5] = sz - dz;
        ef[w][m][6] = 0.0f; ef[w][m][7] = 0.0f;
      }
    }
  }
  __syncthreads();

#pragma unroll
  for (int nb = 0; nb < 64; nb += 16) {
    v8f acc = {0.f, 0.f, 0.f, 0.f, 0.f, 0.f, 0.f, 0.f};
#pragma unroll
    for (int kb = 0; kb < KP; kb += 4) {
      v2f a = *(const v2f*)&ef[w][m][kb + koff];
      v2f b = *(const v2f*)&w1t[nb + m][kb + koff];
      acc = wmma4(a, b, acc);
    }
    const int ch = nb + m;
    const float bb = b1s[ch];
    float sl = 0.f, ql = 0.f;
#pragma unroll
    for (int r = 0; r < 8; ++r) {
      const int row = r + hi * 8;
      if (base + row < E) {
        float v = acc[r] + bb;
        sl += v;
        ql += v * v;
      }
    }
    atomicAdd(&sSum[ch], sl);   // ds_add_f32
    atomicAdd(&sSq[ch], ql);
  }
  __syncthreads();
  if (tid < 64) {
    atomicAdd(&gSum[tid], sSum[tid]);  // global_atomic_add_f32
    atomicAdd(&gSq[tid], sSq[tid]);
  }
}

// ---------------------------------------------------------------------------
// BN finalize: scale = g * rsqrt(var+eps), shift = beta - mu*scale.
// Also resets the accumulators for the next layer.
// ---------------------------------------------------------------------------
__global__ void bn_stats(const float* __restrict__ gamma,
                         const float* __restrict__ beta,
                         float* __restrict__ gSum, float* __restrict__ gSq,
                         float* __restrict__ scsh, float invE) {
  int c = threadIdx.x;
  if (c < 64) {
    float mu = gSum[c] * invE;
    float var = gSq[c] * invE - mu * mu;
    float rs = rsqrtf(var + 1e-5f);
    float sc = gamma[c] * rs;
    scsh[c] = sc;
    scsh[64 + c] = beta[c] - mu * sc;
    gSum[c] = 0.0f;
    gSq[c] = 0.0f;
  }
}

// ---------------------------------------------------------------------------
// Pass C: recompute x (WMMA GEMM1), BN+ReLU -> xn (LDS), GEMM2 (WMMA),
// scatter-max into node features via int atomicMax (all values >= 0).
// ---------------------------------------------------------------------------
template <int KP, int KS, bool SECOND>
__global__ __launch_bounds__(BLOCK) void edge_pass_c(
    const float* __restrict__ pos, const int* __restrict__ esrc,
    const int* __restrict__ edst, const float* __restrict__ hprev,
    const float* __restrict__ W1, const float* __restrict__ b1,
    const float* __restrict__ scsh, const float* __restrict__ W2,
    const float* __restrict__ b2, float* __restrict__ hout, int E) {
  __shared__ float w1t[64][KS];
  __shared__ float w2t[64][68];
  __shared__ float ef[WAVES][16][KS];
  __shared__ float xn[WAVES][16][68];  // normalized+relu'd x tile (A of GEMM2)
  __shared__ float b1s[64], b2s[64], scs[64], shs[64];
  __shared__ int dsts[WAVES][16];

  const int tid = threadIdx.x;
  const int w = tid >> 5;
  const int lane = tid & 31;
  const int m = lane & 15;
  const int hi = lane >> 4;
  const int koff = hi * 2;
  const int KIN = SECOND ? 67 : 6;

  for (int idx = tid; idx < 64 * KS; idx += BLOCK) {
    int n = idx / KS, k = idx - n * KS;
    w1t[n][k] = (k < KIN) ? W1[k * 64 + n] : 0.0f;
  }
  for (int idx = tid; idx < 64 * 68; idx += BLOCK) {
    int n = idx / 68, k = idx - n * 68;
    w2t[n][k] = (k < 64) ? W2[k * 64 + n] : 0.0f;
  }
  if (tid < 64) {
    b1s[tid] = b1[tid];
    b2s[tid] = b2[tid];
    scs[tid] = scsh[tid];
    shs[tid] = scsh[64 + tid];
  }

  const int base = (blockIdx.x * WAVES + w) * 16;
  {
    int e = base + m;
    if (e >= E) e = E - 1;
    int s = esrc[e];
    if (SECOND) {
      const float4* hr = (const float4*)(hprev + (size_t)s * 64);
      float4* d4 = (float4*)&ef[w][m][hi * 32];
#pragma unroll
      for (int i = 0; i < 8; ++i) d4[i] = hr[hi * 8 + i];
    }
    if (lane < 16) {
      int d = edst[e];
      dsts[w][m] = d;
      float sx = pos[3 * s], sy = pos[3 * s + 1], sz = pos[3 * s + 2];
      float dx = pos[3 * d], dy = pos[3 * d + 1], dz = pos[3 * d + 2];
      if (SECOND) {
        ef[w][m][64] = sx - dx; ef[w][m][65] = sy - dy;
        ef[w][m][66] = sz - dz; ef[w][m][67] = 0.0f;
      } else {
        ef[w][m][0] = sx; ef[w][m][1] = sy; ef[w][m][2] = sz;
        ef[w][m][3] = sx - dx; ef[w][m][4] = sy - dy; ef[w][m][5] = sz - dz;
        ef[w][m][6] = 0.0f; ef[w][m][7] = 0.0f;
      }
    }
  }
  __syncthreads();

  // GEMM1 + BN + ReLU -> xn
#pragma unroll
  for (int nb = 0; nb < 64; nb += 16) {
    v8f acc = {0.f, 0.f, 0.f, 0.f, 0.f, 0.f, 0.f, 0.f};
#pragma unroll
    for (int kb = 0; kb < KP; kb += 4) {
      v2f a = *(const v2f*)&ef[w][m][kb + koff];
      v2f b = *(const v2f*)&w1t[nb + m][kb + koff];
      acc = wmma4(a, b, acc);
    }
    const int ch = nb + m;
    const float bb = b1s[ch], sc = scs[ch], sh = shs[ch];
#pragma unroll
    for (int r = 0; r < 8; ++r) {
      const int row = r + hi * 8;
      float v = (acc[r] + bb) * sc + sh;
      xn[w][row][ch] = v > 0.f ? v : 0.f;
    }
  }
  __syncthreads();

  // GEMM2 + bias, scatter-max (relu folded into 0-init + skip of y<=0)
#pragma unroll
  for (int nb = 0; nb < 64; nb += 16) {
    v8f acc = {0.f, 0.f, 0.f, 0.f, 0.f, 0.f, 0.f, 0.f};
#pragma unroll
    for (int kb = 0; kb < 64; kb += 4) {
      v2f a = *(const v2f*)&xn[w][m][kb + koff];
      v2f b = *(const v2f*)&w2t[nb + m][kb + koff];
      acc = wmma4(a, b, acc);
    }
    const int ch = nb + m;
    const float bb = b2s[ch];
#pragma unroll
    for (int r = 0; r < 8; ++r) {
      const int row = r + hi * 8;
      if (base + row < E) {
        float y = acc[r] + bb;
        if (y > 0.f)  // non-negative floats: int bit pattern is monotone
          atomicMax((int*)&hout[(size_t)dsts[w][row] * 64 + ch],
                    __float_as_int(y));
      }
    }
  }
}

// ---------------------------------------------------------------------------
// Global max pool over batch vector (all h >= 0, g pre-zeroed).
// ---------------------------------------------------------------------------
__global__ void pool_kernel(const float* __restrict__ h,
                            const int* __restrict__ batch,
                            float* __restrict__ g, int N) {
  int i = blockIdx.x * blockDim.x + threadIdx.x;
  if (i < N * 64) {
    float v = h[i];
    if (v > 0.f) {
      int n = i >> 6, c = i & 63;
      atomicMax((int*)&g[(size_t)batch[n] * 64 + c], __float_as_int(v));
    }
  }
}

// Final classifier: [G,64] @ [64,C] + bc
__global__ void head_kernel(const float* __restrict__ g,
                            const float* __restrict__ Wc,
                            const float* __restrict__ bc,
                            float* __restrict__ out, int G, int C) {
  int i = blockIdx.x * blockDim.x + threadIdx.x;
  if (i < G * C) {
    int gi = i / C, c = i - gi * C;
    float s = bc[c];
    for (int k = 0; k < 64; ++k) s += g[gi * 64 + k] * Wc[k * C + c];
    out[i] = s;
  }
}

extern "C" void kernel_launch(void* const* d_in, const int* in_sizes, int n_in,
                              void* d_out, int out_size, void* d_ws,
                              size_t ws_size, hipStream_t stream) {
  const float* pos = (const float*)d_in[0];
  const int* ei = (const int*)d_in[1];
  const int* batch = (const int*)d_in[2];
  const float* W1a = (const float*)d_in[3];
  const float* b1a = (const float*)d_in[4];
  const float* g1a = (const float*)d_in[5];
  const float* be1a = (const float*)d_in[6];
  const float* W2a = (const float*)d_in[7];
  const float* b2a = (const float*)d_in[8];
  const float* W1b = (const float*)d_in[9];
  const float* b1b = (const float*)d_in[10];
  const float* g1b = (const float*)d_in[11];
  const float* be1b = (const float*)d_in[12];
  const float* W2b = (const float*)d_in[13];
  const float* b2b = (const float*)d_in[14];
  const float* Wc = (const float*)d_in[15];
  const float* bc = (const float*)d_in[16];

  const int N = in_sizes[0] / 3;
  const int E = in_sizes[1] / 2;
  const int G = 64;
  const int C = in_sizes[16];
  const int* esrc = ei;
  const int* edst = ei + E;

  float* ws = (float*)d_ws;
  float* h1 = ws;                        // [N,64]
  float* h2 = h1 + (size_t)N * 64;       // [N,64]
  float* gbuf = h2 + (size_t)N * 64;     // [G,64]
  float* gSum = gbuf + (size_t)G * 64;   // [64]
  float* gSq = gSum + 64;                // [64]
  float* scsh = gSq + 64;                // [128]
  size_t totalFloats = (size_t)2 * N * 64 + (size_t)G * 64 + 256;
  hipMemsetAsync(d_ws, 0, totalFloats * sizeof(float), stream);

  const int nTiles = (E + 15) / 16;
  const int nBlocksE = (nTiles + WAVES - 1) / WAVES;
  const float invE = 1.0f / (float)E;

  // Layer 1 (in = 6, K padded to 8, ef stride 12 for bank-conflict-free frags)
  edge_pass_a<8, 12, false><<<nBlocksE, BLOCK, 0, stream>>>(
      pos, esrc, edst, nullptr, W1a, b1a, gSum, gSq, E);
  bn_stats<<<1, 64, 0, stream>>>(g1a, be1a, gSum, gSq, scsh, invE);
  edge_pass_c<8, 12, false><<<nBlocksE, BLOCK, 0, stream>>>(
      pos, esrc, edst, nullptr, W1a, b1a, scsh, W2a, b2a, h1, E);

  // Layer 2 (in = 67, K padded to 68)
  edge_pass_a<68, 68, true><<<nBlocksE, BLOCK, 0, stream>>>(
      pos, esrc, edst, h1, W1b, b1b, gSum, gSq, E);
  bn_stats<<<1, 64, 0, stream>>>(g1b, be1b, gSum, gSq, scsh, invE);
  edge_pass_c<68, 68, true><<<nBlocksE, BLOCK, 0, stream>>>(
      pos, esrc, edst, h1, W1b, b1b, scsh, W2b, b2b, h2, E);

  // Global max pool + classifier head
  pool_kernel<<<(int)(((size_t)N * 64 + 255) / 256), 256, 0, stream>>>(
      h2, batch, gbuf, N);
  head_kernel<<<(G * C + 255) / 256, 256, 0, stream>>>(gbuf, Wc, bc,
                                                       (float*)d_out, G, C);
}